// GNN_33569464385600
// MI455X (gfx1250) — compile-verified
//
#include <hip/hip_runtime.h>

typedef __attribute__((ext_vector_type(2))) float v2f;
typedef __attribute__((ext_vector_type(8))) float v8f;

#define D 128

// ---------------------------------------------------------------------------
// Zero-fill
// ---------------------------------------------------------------------------
__global__ __launch_bounds__(256) void gnn_zero(float* __restrict__ p, long long n) {
    long long i = (long long)blockIdx.x * blockDim.x + threadIdx.x;
    if (i < n) p[i] = 0.0f;
}

// ---------------------------------------------------------------------------
// Degree count: cnt[dst[e]] += 1
// ---------------------------------------------------------------------------
__global__ __launch_bounds__(256) void gnn_count(const int* __restrict__ dst,
                                                 float* __restrict__ cnt, int nEdges) {
    int e = blockIdx.x * blockDim.x + threadIdx.x;
    if (e < nEdges) atomicAdd(&cnt[dst[e]], 1.0f);
}

// inv[n] = 1 / max(cnt[n], 1)   (in place)
__global__ __launch_bounds__(256) void gnn_inv(float* __restrict__ cnt, int nNodes) {
    int n = blockIdx.x * blockDim.x + threadIdx.x;
    if (n < nNodes) cnt[n] = 1.0f / fmaxf(cnt[n], 1.0f);
}

// ---------------------------------------------------------------------------
// Edge scatter-add: agg[dst[e], :] += feat[src[e], :]
// 32 lanes per edge, each lane handles 4 consecutive columns (float4 load,
// 4x global_atomic_add_f32). All traffic is L2-resident (25.6MB node table).
// ---------------------------------------------------------------------------
__global__ __launch_bounds__(256) void gnn_scatter(const float* __restrict__ feat,
                                                   const int* __restrict__ src,
                                                   const int* __restrict__ dst,
                                                   float* __restrict__ agg, int nEdges) {
    long long t = (long long)blockIdx.x * blockDim.x + threadIdx.x;
    int e = (int)(t >> 5);
    int c = ((int)t & 31) * 4;
    if (e >= nEdges) return;
    int s = src[e];
    int d = dst[e];
    float4 val = *(const float4*)(feat + (size_t)s * D + c);
    float* p = agg + (size_t)d * D + c;
    atomicAdd(p + 0, val.x);
    atomicAdd(p + 1, val.y);
    atomicAdd(p + 2, val.z);
    atomicAdd(p + 3, val.w);
}

// ---------------------------------------------------------------------------
// Fused SAGE linear:  out = act( (agg*inv) @ wl^T + xin @ wr^T + bias )
//
// One wave -> one 16x16 output tile (v_wmma_f32_16x16x4_f32, K stepped by 4).
// Block = 256 threads = 8 waves = one 16-node row-band x all 128 columns.
// Grid  = N/16 blocks (N = 50000 = 3125*16, exact tiling -> EXEC all ones).
//
// fp32 A 16x4 layout (per ISA): lane L holds float2 {A[L&15, k0], A[L&15, k0+1]}
// with k0 = kc + 2*(L>>4). B 4x16 mirrors it with rows of W (row-major W gives
// the @W^T the reference needs). C/D: lane L, vgpr v -> (M = v + 8*(L>>4),
// N = L&15).
// ---------------------------------------------------------------------------
__global__ __launch_bounds__(256) void gnn_sage_gemm(
    const float* __restrict__ agg,   // [N, D] scatter sums
    const float* __restrict__ inv,   // [N]    1/max(deg,1)
    const float* __restrict__ xin,   // [N, D] self features
    const float* __restrict__ wl,    // [D, D] row-major
    const float* __restrict__ wr,    // [D, D] row-major
    const float* __restrict__ bias,  // [D]
    float* __restrict__ out,         // [N, D]
    int relu) {
    const int lane    = threadIdx.x & 31;
    const int wave    = threadIdx.x >> 5;       // 0..7 column tile
    const int rowBase = blockIdx.x * 16;
    const int colBase = wave * 16;

    const int half = lane >> 4;                 // 0 or 1
    const int l15  = lane & 15;
    const int arow = rowBase + l15;             // node row for A operands
    const int brow = colBase + l15;             // weight row (output column j)

    const float inv_c = inv[arow];
    const float* aggRow = agg + (size_t)arow * D;
    const float* xRow   = xin + (size_t)arow * D;
    const float* wlRow  = wl + (size_t)brow * D;
    const float* wrRow  = wr + (size_t)brow * D;

    v8f acc = {};
#pragma unroll 8
    for (int kc = 0; kc < D; kc += 4) {
        const int k0 = kc + 2 * half;
        v2f a_m = *(const v2f*)(aggRow + k0);
        a_m.x *= inv_c;
        a_m.y *= inv_c;
        v2f a_x = *(const v2f*)(xRow + k0);
        v2f b_l = *(const v2f*)(wlRow + k0);
        v2f b_r = *(const v2f*)(wrRow + k0);
        acc = __builtin_amdgcn_wmma_f32_16x16x4_f32(false, a_m, false, b_l,
                                                    (short)0, acc, false, false);
        acc = __builtin_amdgcn_wmma_f32_16x16x4_f32(false, a_x, false, b_r,
                                                    (short)0, acc, false, false);
    }

    const float bv = bias[colBase + l15];
#pragma unroll
    for (int v = 0; v < 8; ++v) {
        float r = acc[v] + bv;
        if (relu) r = fmaxf(r, 0.0f);
        out[(size_t)(rowBase + v + 8 * half) * D + colBase + l15] = r;
    }
}

// ---------------------------------------------------------------------------
// Launch
// ---------------------------------------------------------------------------
extern "C" void kernel_launch(void* const* d_in, const int* in_sizes, int n_in,
                              void* d_out, int out_size, void* d_ws, size_t ws_size,
                              hipStream_t stream) {
    const float* x   = (const float*)d_in[0];
    const int* edge  = (const int*)d_in[1];
    const float* w1l = (const float*)d_in[2];
    const float* w1r = (const float*)d_in[3];
    const float* b1  = (const float*)d_in[4];
    const float* w2l = (const float*)d_in[5];
    const float* w2r = (const float*)d_in[6];
    const float* b2  = (const float*)d_in[7];
    float* out = (float*)d_out;

    const int N = in_sizes[0] / D;      // 50000
    const int E = in_sizes[1] / 2;      // 800000
    const int* src = edge;
    const int* dst = edge + E;

    // workspace: agg [N*D] | h [N*D] | inv [N]
    float* agg = (float*)d_ws;
    float* h   = agg + (size_t)N * D;
    float* inv = h + (size_t)N * D;

    const long long nd = (long long)N * D;
    dim3 blk(256);
    dim3 gZeroND((unsigned)((nd + 255) / 256));
    dim3 gZeroN((unsigned)((N + 255) / 256));
    dim3 gCnt((unsigned)((E + 255) / 256));
    dim3 gScat((unsigned)(((long long)E * 32 + 255) / 256));
    dim3 gGemm((unsigned)(N / 16));     // N = 50000 tiles exactly

    // degree (shared by both layers)
    gnn_zero<<<gZeroN, blk, 0, stream>>>(inv, N);
    gnn_count<<<gCnt, blk, 0, stream>>>(dst, inv, E);
    gnn_inv<<<gZeroN, blk, 0, stream>>>(inv, N);

    // layer 1
    gnn_zero<<<gZeroND, blk, 0, stream>>>(agg, nd);
    gnn_scatter<<<gScat, blk, 0, stream>>>(x, src, dst, agg, E);
    gnn_sage_gemm<<<gGemm, blk, 0, stream>>>(agg, inv, x, w1l, w1r, b1, h, 1);

    // layer 2
    gnn_zero<<<gZeroND, blk, 0, stream>>>(agg, nd);
    gnn_scatter<<<gScat, blk, 0, stream>>>(h, src, dst, agg, E);
    gnn_sage_gemm<<<gGemm, blk, 0, stream>>>(agg, inv, h, w2l, w2r, b2, out, 0);
}